// GlobalRouters_88124138979396
// MI455X (gfx1250) — compile-verified
//
#include <hip/hip_runtime.h>
#include <hip/hip_bf16.h>
#include <math.h>

typedef float v2f __attribute__((ext_vector_type(2)));
typedef float v8f __attribute__((ext_vector_type(8)));

#define D_MODEL 1024
#define NSEG    512        // neurons per segment (softmax width)
#define DSP     64         // d_space
#define NCOL    384        // 6 * d_space
#define MROWS   8192       // B * S
#define TOPK    8
#define EPSF    1e-8f
#define NEG_INF (-3.402823466e+38f)

// ---------------------------------------------------------------------------
// Kernel 0: L2-normalize neuron embeddings. One wave32 per 64-wide row.
// ---------------------------------------------------------------------------
__global__ void GR_normalize_emb(const float* __restrict__ emb,
                                 float* __restrict__ out, int rows) {
    int wave = (int)((blockIdx.x * blockDim.x + threadIdx.x) >> 5);
    int lane = threadIdx.x & 31;
    if (wave >= rows) return;
    const float* r = emb + (size_t)wave * DSP;
    float a0 = r[lane];
    float a1 = r[lane + 32];
    float ss = a0 * a0 + a1 * a1;
    #pragma unroll
    for (int m = 16; m >= 1; m >>= 1) ss += __shfl_xor(ss, m);
    float inv = 1.0f / (sqrtf(ss) + EPSF);
    float* o = out + (size_t)wave * DSP;
    o[lane]      = a0 * inv;
    o[lane + 32] = a1 * inv;
}

// ---------------------------------------------------------------------------
// Kernel 1: proj = x @ W_all + b_all   (8192x1024)·(1024x384)
// 4 waves / block; each wave owns one 16x16 output tile, K swept in steps of 4
// via V_WMMA_F32_16X16X4_F32. Two accumulator chains for ILP.
//
// f32 WMMA lane layout (ISA 7.12.2):
//   A 16x4 : lane L holds row M=L&15; VGPR0/1 = K=(L>=16?2:0)+{0,1}
//   B 4x16 : lane L holds col N=L&15; VGPR0/1 = K=(L>=16?2:0)+{0,1}
//   C 16x16: VGPR r = row (r + 8*(L>>4)), col L&15
// ---------------------------------------------------------------------------
__global__ __launch_bounds__(128)
void GR_proj_gemm(const float* __restrict__ x, const float* __restrict__ W,
                  const float* __restrict__ bias, float* __restrict__ proj) {
    int wave  = threadIdx.x >> 5;
    int lane  = threadIdx.x & 31;
    int tileM = blockIdx.x * 16;
    int tileN = (blockIdx.y * 4 + wave) * 16;   // 24 N-tiles total, 4 per block
    int mn    = lane & 15;                      // M for A-frag, N for B-frag
    int kk    = (lane >> 4) << 1;               // 0 or 2

    const float* xrow = x + (size_t)(tileM + mn) * D_MODEL;
    const float* wcol = W + tileN + mn;

    v8f c0 = {};
    v8f c1 = {};
    for (int k = 0; k < D_MODEL; k += 8) {
        v2f a0, b0, a1, b1;
        a0.x = xrow[k + kk];
        a0.y = xrow[k + kk + 1];
        b0.x = wcol[(size_t)(k + kk) * NCOL];
        b0.y = wcol[(size_t)(k + kk + 1) * NCOL];
        a1.x = xrow[k + 4 + kk];
        a1.y = xrow[k + 4 + kk + 1];
        b1.x = wcol[(size_t)(k + 4 + kk) * NCOL];
        b1.y = wcol[(size_t)(k + 4 + kk + 1) * NCOL];
        c0 = __builtin_amdgcn_wmma_f32_16x16x4_f32(false, a0, false, b0,
                                                   (short)0, c0, false, false);
        c1 = __builtin_amdgcn_wmma_f32_16x16x4_f32(false, a1, false, b1,
                                                   (short)0, c1, false, false);
    }
    float bv = bias[tileN + mn];
    int hi = lane >> 4;
    float* o = proj + (size_t)(tileM + hi * 8) * NCOL + tileN + mn;
    #pragma unroll
    for (int r = 0; r < 8; ++r)
        o[(size_t)r * NCOL] = c0[r] + c1[r] + bv;
}

// ---------------------------------------------------------------------------
// Kernel 2 (fused): per (head, 16-row block):
//   logits[16x512] = h_tile @ emb_seg^T   (WMMA f32, staged to LDS)
//   then softmax stats + iterative top-8 argmax + renormalized sparse scatter.
// Block = 256 threads = 8 waves; wave w computes columns [w*64, w*64+64).
// ---------------------------------------------------------------------------
__global__ __launch_bounds__(256)
void GR_head_attend(const float* __restrict__ proj,
                    const float* __restrict__ embn,
                    float* __restrict__ out) {
    __shared__ float lds[16 * NSEG];            // 32 KB logits tile

    const int segStart[6] = {0, 0, NSEG, 2 * NSEG, 2 * NSEG, 3 * NSEG};
    int head    = blockIdx.y;
    int rowBase = blockIdx.x * 16;
    int tid  = threadIdx.x;
    int wave = tid >> 5;
    int lane = tid & 31;
    int mn   = lane & 15;
    int kk   = (lane >> 4) << 1;

    // ---- WMMA phase: 4 independent 16x16 tiles per wave (cols n0..n0+63) ----
    const float* A = proj + (size_t)(rowBase + mn) * NCOL + head * DSP;
    const float* E = embn + (size_t)segStart[head] * DSP;
    int n0 = wave * 64;
    const float* B0 = E + (size_t)(n0 +  0 + mn) * DSP;
    const float* B1 = E + (size_t)(n0 + 16 + mn) * DSP;
    const float* B2 = E + (size_t)(n0 + 32 + mn) * DSP;
    const float* B3 = E + (size_t)(n0 + 48 + mn) * DSP;

    v8f c0 = {}, c1 = {}, c2 = {}, c3 = {};
    for (int k = 0; k < DSP; k += 4) {
        v2f a, b0, b1, b2, b3;
        a.x  = A[k + kk];      a.y  = A[k + kk + 1];
        b0.x = B0[k + kk];     b0.y = B0[k + kk + 1];
        b1.x = B1[k + kk];     b1.y = B1[k + kk + 1];
        b2.x = B2[k + kk];     b2.y = B2[k + kk + 1];
        b3.x = B3[k + kk];     b3.y = B3[k + kk + 1];
        c0 = __builtin_amdgcn_wmma_f32_16x16x4_f32(false, a, false, b0,
                                                   (short)0, c0, false, false);
        c1 = __builtin_amdgcn_wmma_f32_16x16x4_f32(false, a, false, b1,
                                                   (short)0, c1, false, false);
        c2 = __builtin_amdgcn_wmma_f32_16x16x4_f32(false, a, false, b2,
                                                   (short)0, c2, false, false);
        c3 = __builtin_amdgcn_wmma_f32_16x16x4_f32(false, a, false, b3,
                                                   (short)0, c3, false, false);
    }
    int hi = lane >> 4;
    #pragma unroll
    for (int r = 0; r < 8; ++r) {
        int row = r + hi * 8;
        lds[row * NSEG + n0 +  0 + mn] = c0[r];
        lds[row * NSEG + n0 + 16 + mn] = c1[r];
        lds[row * NSEG + n0 + 32 + mn] = c2[r];
        lds[row * NSEG + n0 + 48 + mn] = c3[r];
    }
    __syncthreads();

    // ---- softmax + top-8 phase: 16 threads per row (one half-wave each) ----
    int row = tid >> 4;       // rows 2w,2w+1 live in lanes [0..15]/[16..31] of wave w
    int sub = tid & 15;       // 16 threads per row; xor masks 1/2/4/8 stay in-half
    const float* lrow = lds + row * NSEG;

    float vloc[32];
    #pragma unroll
    for (int j = 0; j < 32; ++j) vloc[j] = lrow[sub * 32 + j];

    // row max
    float rmax = NEG_INF;
    #pragma unroll
    for (int j = 0; j < 32; ++j) rmax = fmaxf(rmax, vloc[j]);
    #pragma unroll
    for (int s = 1; s <= 8; s <<= 1) rmax = fmaxf(rmax, __shfl_xor(rmax, s));

    // full softmax denominator Z (needed only for the +EPS term)
    float z = 0.0f;
    #pragma unroll
    for (int j = 0; j < 32; ++j) z += __expf(vloc[j] - rmax);
    #pragma unroll
    for (int s = 1; s <= 8; s <<= 1) z += __shfl_xor(z, s);

    // iterative top-8 argmax (tie -> lower index, matching lax.top_k order)
    unsigned used = 0u;
    int   selIdx[TOPK];
    float selEv[TOPK];
    float T = 0.0f;
    #pragma unroll
    for (int it = 0; it < TOPK; ++it) {
        float bv = NEG_INF;
        int   bi = 0x7fffffff;
        #pragma unroll
        for (int j = 0; j < 32; ++j) {
            if (!((used >> j) & 1u)) {
                float v = vloc[j];
                int  gi = sub * 32 + j;
                if (v > bv || (v == bv && gi < bi)) { bv = v; bi = gi; }
            }
        }
        #pragma unroll
        for (int s = 1; s <= 8; s <<= 1) {
            float ov = __shfl_xor(bv, s);
            int   oi = __shfl_xor(bi, s);
            if (ov > bv || (ov == bv && oi < bi)) { bv = ov; bi = oi; }
        }
        if ((bi >> 5) == sub) used |= 1u << (bi & 31);   // owner retires it
        float ev = __expf(bv - rmax);
        T += ev;
        selIdx[it] = bi;
        selEv[it]  = ev;
    }
    // out_i = e_i / (T + EPS*Z)  [== (e_i/Z) / (T/Z + EPS)]
    float inv = 1.0f / (T + EPSF * z);

    float* orow = out + ((size_t)head * MROWS + rowBase + row) * NSEG;
    #pragma unroll
    for (int j = 0; j < 32; ++j) {
        int gi = sub * 32 + j;
        float val = 0.0f;
        #pragma unroll
        for (int it = 0; it < TOPK; ++it)
            if (gi == selIdx[it]) val = selEv[it] * inv;
        orow[gi] = val;
    }
}

// ---------------------------------------------------------------------------
extern "C" void kernel_launch(void* const* d_in, const int* in_sizes, int n_in,
                              void* d_out, int out_size, void* d_ws, size_t ws_size,
                              hipStream_t stream) {
    const float* x     = (const float*)d_in[0];   // [4,2048,1024]
    const float* W_all = (const float*)d_in[1];   // [1024,384]
    const float* b_all = (const float*)d_in[2];   // [384]
    const float* emb   = (const float*)d_in[3];   // [3072,64]
    float* out = (float*)d_out;                   // [6,4,2048,512]

    // workspace layout: embn (3072*64 f32) | proj (8192*384 f32)  ~13.4 MB
    float* embn = (float*)d_ws;
    float* proj = embn + 3072 * DSP;

    // 3072 rows, one wave each -> 384 blocks of 256 threads
    GR_normalize_emb<<<dim3((3072 * 32 + 255) / 256), 256, 0, stream>>>(emb, embn, 3072);

    // M tiles = 8192/16 = 512; N tiles = 384/16 = 24 -> 6 blocks of 4 waves
    GR_proj_gemm<<<dim3(512, 6), 128, 0, stream>>>(x, W_all, b_all, proj);

    // 512 row-blocks x 6 heads; 256 threads (8 waves) each
    GR_head_attend<<<dim3(512, 6), 256, 0, stream>>>(proj, embn, out);
}